// RMSE_high_Loss_50294067036177
// MI455X (gfx1250) — compile-verified
//
#include <hip/hip_runtime.h>
#include <math.h>

// ---------------------------------------------------------------------------
// Per-channel weighted RMSE, summed over channels.
//   shapes: [NT=1200, NB=16384, NY=8] f32
//   w[t] = 1.75 if (t%12) in [6,9] else 0.2
//   loss = sum_y sqrt( mean_{t,b} (w[t]*(o-t))^2 )
//
// Memory-bound: 1.26 GB read -> ~54 us floor at 23.3 TB/s.
// Kernel 1: vectorized b128 streaming + register accumulation + deterministic
//           LDS tree reduction -> 8 per-block partial sums in d_ws.
// Kernel 2: single wave32 reduces the [nblocks][8] partials with
//           V_WMMA_F32_16X16X4_F32 against an all-ones B matrix (rows =
//           channels, K = 4 blocks/instr, C accumulates), then sqrt+sum.
// ---------------------------------------------------------------------------

#define NT 1200
#define NB 16384
#define NY 8
#define W_LOW2  (0.2f * 0.2f)
#define W_HIGH2 (1.75f * 1.75f)
#define V4_PER_T ((NB * NY) / 4)   // 32768 float4 per t-slice

typedef __attribute__((ext_vector_type(2))) float v2f;
typedef __attribute__((ext_vector_type(8))) float v8f;

__global__ void rmse_partial_kernel(const float4* __restrict__ out,
                                    const float4* __restrict__ tgt,
                                    float* __restrict__ partials,
                                    int total_v4) {
    const int tid = threadIdx.x;
    const int g = blockIdx.x * blockDim.x + tid;
    const int stride = gridDim.x * blockDim.x;   // even -> parity fixed per thread

    // Each float4 at index v covers channels 4*(v&1) .. 4*(v&1)+3.
    float a0 = 0.f, a1 = 0.f, a2 = 0.f, a3 = 0.f;

    for (int v = g; v < total_v4; v += stride) {
        const int t = v >> 15;            // v / V4_PER_T
        const int mon = t % 12;
        const float w2 = (mon >= 6 && mon <= 9) ? W_HIGH2 : W_LOW2;

        const float4 ov = out[v];         // global_load_b128
        const float4 tv = tgt[v];         // global_load_b128
        const float d0 = ov.x - tv.x;
        const float d1 = ov.y - tv.y;
        const float d2 = ov.z - tv.z;
        const float d3 = ov.w - tv.w;
        a0 = fmaf(w2 * d0, d0, a0);
        a1 = fmaf(w2 * d1, d1, a1);
        a2 = fmaf(w2 * d2, d2, a2);
        a3 = fmaf(w2 * d3, d3, a3);
    }

    // Deterministic block reduction: fixed-order sums only (no float atomics,
    // so the graph-replayed result is bitwise identical every call).
    __shared__ float sdata[256 * 4];
    __shared__ float saux[64];
    sdata[tid * 4 + 0] = a0;
    sdata[tid * 4 + 1] = a1;
    sdata[tid * 4 + 2] = a2;
    sdata[tid * 4 + 3] = a3;
    __syncthreads();

    // Stage 1: 8 threads per channel, each sums 16 of the 128 contributors.
    if (tid < 64) {
        const int c = tid >> 3;          // channel 0..7
        const int i = tid & 7;           // sub-chunk 0..7
        const int p = c >> 2;            // thread parity owning this channel
        const int j = c & 3;             // component within float4
        float s = 0.f;
        for (int q = 0; q < 16; ++q) {
            const int th = 2 * (i * 16 + q) + p;   // 0..255
            s += sdata[th * 4 + j];
        }
        saux[tid] = s;                   // saux[c*8 + i]
    }
    __syncthreads();

    // Stage 2: 8 threads finish the 8 channels.
    if (tid < 8) {
        float s = 0.f;
        for (int i = 0; i < 8; ++i) s += saux[tid * 8 + i];
        partials[blockIdx.x * 8 + tid] = s;
    }
}

// One wave32. Reduces partials[nblocks][8] along the block axis using
// V_WMMA_F32_16X16X4_F32 with an all-ones B: D[m][n] = sum_k A[m][k] + C.
// A rows m = channels (rows 8..15 zero); per ISA layout lanes 0-15 supply
// A[m][0..1] (m = lane) and lanes 16-31 supply A[m][2..3] (m = lane-16).
// After the loop, C VGPR j on lane 0 holds the total for channel j.
__global__ void rmse_finalize_kernel(const float* __restrict__ partials,
                                     float* __restrict__ loss_out,
                                     int nblocks) {
    const int lane = threadIdx.x;        // 0..31
    const int half = lane >> 4;          // 0: k={0,1}, 1: k={2,3}
    const int row  = lane & 15;          // A-matrix row m

    v8f c = {};
    const v2f ones = {1.0f, 1.0f};

    for (int base = 0; base < nblocks; base += 4) {
        float v0 = 0.f, v1 = 0.f;
        if (row < 8) {                   // channels live in rows 0..7
            v0 = partials[(base + 2 * half + 0) * 8 + row];
            v1 = partials[(base + 2 * half + 1) * 8 + row];
        }
        v2f a;
        a[0] = v0;
        a[1] = v1;
        // v_wmma_f32_16x16x4_f32: D = A x Ones + C  (row-sum accumulate)
        c = __builtin_amdgcn_wmma_f32_16x16x4_f32(
                /*neg_a=*/false, a, /*neg_b=*/false, ones,
                /*c_mod=*/(short)0, c, /*reuse_a=*/false, /*reuse_b=*/false);
    }

    if (lane == 0) {
        const float inv_n = 1.0f / ((float)NT * (float)NB);
        float loss = 0.f;
        for (int m = 0; m < 8; ++m) loss += sqrtf(c[m] * inv_n);
        loss_out[0] = loss;
    }
}

extern "C" void kernel_launch(void* const* d_in, const int* in_sizes, int n_in,
                              void* d_out, int out_size, void* d_ws, size_t ws_size,
                              hipStream_t stream) {
    const float4* out4 = (const float4*)d_in[0];
    const float4* tgt4 = (const float4*)d_in[1];
    float* loss = (float*)d_out;
    float* partials = (float*)d_ws;

    int nblocks = 1024;                              // 32 KB of partials
    if (ws_size < (size_t)nblocks * 8 * sizeof(float)) {
        nblocks = (int)(ws_size / (8 * sizeof(float))) & ~3;  // multiple of 4
        if (nblocks < 4) nblocks = 4;
    }

    const int total_v4 = (NT * NB * NY) / 4;         // 39,321,600

    rmse_partial_kernel<<<nblocks, 256, 0, stream>>>(out4, tgt4, partials, total_v4);
    rmse_finalize_kernel<<<1, 32, 0, stream>>>(partials, loss, nblocks);
}